// Backbone_4449586118747
// MI455X (gfx1250) — compile-verified
//
#include <hip/hip_runtime.h>
#include <math.h>

// ---------------------------------------------------------------------------
// Problem constants (from reference: B=32, L=512, D=321, P=16, pred=96)
// ---------------------------------------------------------------------------
#define PLEN   16
#define LSEQ   512
#define NB     32
#define NVAR   321
#define NPAT   32
#define PRED   96
#define BD     (NB * NVAR)          /* 10272 */
#define STOT   (BD * LSEQ)          /* 5259264 */
#define EPSBN  1e-5f

typedef __attribute__((ext_vector_type(2))) float v2f;
typedef __attribute__((ext_vector_type(8))) float v8f;
typedef unsigned int v4u __attribute__((ext_vector_type(4)));
typedef int          v4i __attribute__((ext_vector_type(4)));
typedef int          v8i __attribute__((ext_vector_type(8)));

__device__ __forceinline__ float geluf(float x) {
    return 0.5f * x * (1.0f + erff(x * 0.70710678118654752440f));
}
__device__ __forceinline__ float bn_d(float v, int d, const float* g, const float* b,
                                      const float* m, const float* var) {
    return (v - m[d]) * (g[d] * rsqrtf(var[d] + EPSBN)) + b[d];
}

// ---------------------------------------------------------------------------
// Tensor Data Mover staging (CDNA5): DMA a 32x32 f32 tile from global memory
// into LDS with hardware row padding (pad_interval=32 dwords, pad_amount=1
// dword) -> LDS pitch of 33 floats, i.e. the bank-conflict-free layout the
// WMMA loop consumes.  Falls back to cooperative VALU loads when the builtin
// is unavailable (e.g. host compilation pass).
// ---------------------------------------------------------------------------
#if defined(__gfx1250__) && __has_builtin(__builtin_amdgcn_tensor_load_to_lds) && \
    __has_builtin(__builtin_amdgcn_s_wait_tensorcnt)
#define USE_TDM 1
#else
#define USE_TDM 0
#endif

#if USE_TDM
__device__ __forceinline__ void tdm_load_tile32x32(unsigned lds_byte_off,
                                                   const float* __restrict__ gsrc,
                                                   unsigned ld /* row stride, elems */)
{
    unsigned long long ga = (unsigned long long)(size_t)gsrc;
    v4u g0;
    g0.x = 1u;                                           // count=1, user descriptor
    g0.y = lds_byte_off;                                 // LDS address (bytes)
    g0.z = (unsigned)(ga & 0xffffffffu);                 // global_addr[31:0]
    g0.w = (unsigned)((ga >> 32) & 0x01ffffffu)          // global_addr[56:32]
           | (2u << 30);                                 // type=2 ("image")
    v8i g1;
    g1[0] = (int)((2u << 16)        // data_size = 4 bytes
                | (1u << 20)        // pad_enable
                | (4u << 22)        // pad_interval: 32 dwords
                | (0u << 25));      // pad_amount: 1 dword
    g1[1] = (int)((ld & 0xffffu) << 16);                 // tensor_dim0[15:0]
    g1[2] = (int)((ld >> 16) | (64u << 16));             // dim0[31:16] | tensor_dim1[15:0]=64
    g1[3] = (int)(32u << 16);                            // dim1[31:16]=0 | tile_dim0=32
    g1[4] = (int)(32u);                                  // tile_dim1=32 | tile_dim2=0
    g1[5] = (int)ld;                                     // tensor_dim0_stride[31:0]
    g1[6] = 0;                                           // stride0[47:32] | stride1[15:0]
    g1[7] = 0;                                           // stride1[47:16]
    v4i z4 = {0, 0, 0, 0};
#if __clang_major__ >= 23
    v8i z8 = {0, 0, 0, 0, 0, 0, 0, 0};
    __builtin_amdgcn_tensor_load_to_lds(g0, g1, z4, z4, z8, 0);
#else
    __builtin_amdgcn_tensor_load_to_lds(g0, g1, z4, z4, 0);
#endif
}
#endif

// ---------------------------------------------------------------------------
// WMMA fp32 GEMM: C[M,N] = A[M,K] @ B[K,N] (+ fused epilogue).
// Block = 128 threads = 4 wave32, 2x2 arrangement of 16x16 WMMA tiles
// -> 32x32 C tile per block.  K consumed in chunks of 32; tiles staged into
// LDS by the Tensor Data Mover (TENSORcnt) when available.
// Requires M,N,K multiples of 32 (true for all shapes here).
// modes: 0 raw | 1 +c1w[d]*aux+c1b[d] | 2 conv1(bnD(gelu(acc)))
//        3 +colbias[n] | 4 gelu(acc+colbias[n])
//        5 acc+colbias[n]+aux -> transposed store into [B,pred,D]
// ---------------------------------------------------------------------------
#define TKC 32
#define TILE_FLOATS (32 * 33)                 /* 1056 floats = 4224 bytes */
#define AS(r, c) smem[(r) * 33 + (c)]
#define BS(r, c) smem[TILE_FLOATS + (r) * 33 + (c)]

__global__ __launch_bounds__(128)
void wmma_gemm_f32(const float* __restrict__ A, const float* __restrict__ Bm,
                   float* __restrict__ C, int K, int lda, int ldb, int ldc,
                   int mode,
                   const float* __restrict__ aux,
                   const float* __restrict__ p0, const float* __restrict__ p1,
                   const float* __restrict__ dng, const float* __restrict__ dnb,
                   const float* __restrict__ dnm, const float* __restrict__ dnv)
{
    __shared__ float smem[2 * TILE_FLOATS];   // A tile @ byte 0, B tile @ byte 4224

    const int tid  = threadIdx.x;
    const int wave = tid >> 5;
    const int lane = tid & 31;
    const int wm   = wave >> 1;          // 0..1 (M direction)
    const int wn   = wave & 1;           // 0..1 (N direction)
    const int bm0  = blockIdx.x * 32;
    const int bn0  = blockIdx.y * 32;

    v8f acc = {0.f, 0.f, 0.f, 0.f, 0.f, 0.f, 0.f, 0.f};

    for (int k0 = 0; k0 < K; k0 += TKC) {
#if USE_TDM
        if (tid < 32) {                  // wave 0 drives the TDM and waits on it
            tdm_load_tile32x32(0u, A + (size_t)bm0 * lda + k0, (unsigned)lda);
            tdm_load_tile32x32(TILE_FLOATS * 4u, Bm + (size_t)k0 * ldb + bn0,
                               (unsigned)ldb);
            __builtin_amdgcn_s_wait_tensorcnt((short)0);
        }
        __syncthreads();
        // The TDM writes LDS through a descriptor field the compiler cannot
        // alias-analyze; without this escape+clobber it sees no store to smem
        // and folds the ds_loads below to undef.  Compiles to 0 instructions.
        asm volatile("" : : "v"((const float*)smem) : "memory");
#else
        for (int i = tid; i < 32 * TKC; i += 128) {
            int r = i >> 5, c = i & 31;
            AS(r, c) = A[(size_t)(bm0 + r) * lda + (k0 + c)];
        }
        for (int i = tid; i < TKC * 32; i += 128) {
            int r = i >> 5, c = i & 31;
            BS(r, c) = Bm[(size_t)(k0 + r) * ldb + (bn0 + c)];
        }
        __syncthreads();
#endif
        const int l16  = lane & 15;
        const int koff = (lane >> 4) << 1;   // lanes 16..31 hold K+2,K+3
#pragma unroll
        for (int kk = 0; kk < TKC; kk += 4) {
            v2f a, b;
            a.x = AS(wm * 16 + l16, kk + koff);
            a.y = AS(wm * 16 + l16, kk + koff + 1);
            b.x = BS(kk + koff, wn * 16 + l16);
            b.y = BS(kk + koff + 1, wn * 16 + l16);
            acc = __builtin_amdgcn_wmma_f32_16x16x4_f32(
                false, a, false, b, (short)0, acc, false, false);
        }
        __syncthreads();
    }

    const int gn    = bn0 + wn * 16 + (lane & 15);
    const int rbase = bm0 + wm * 16 + ((lane >> 4) << 3);
#pragma unroll
    for (int r = 0; r < 8; ++r) {
        const int gm = rbase + r;
        const int d  = gm % NVAR;
        float v = acc[r];
        if (mode == 0) {
            C[(size_t)gm * ldc + gn] = v;
        } else if (mode == 1) {          // idct + freq residual (conv1 of xt)
            C[(size_t)gm * ldc + gn] = v + p0[d] * aux[(size_t)gm * ldc + gn] + p1[d];
        } else if (mode == 2) {          // conv1(bnD(gelu(acc)))
            float g  = geluf(v);
            float bn = (g - dnm[d]) * (dng[d] * rsqrtf(dnv[d] + EPSBN)) + dnb[d];
            C[(size_t)gm * ldc + gn] = p0[d] * bn + p1[d];
        } else if (mode == 3) {          // + column bias
            C[(size_t)gm * ldc + gn] = v + p0[gn];
        } else if (mode == 4) {          // gelu(acc + column bias)
            C[(size_t)gm * ldc + gn] = geluf(v + p0[gn]);
        } else {                          // 5: final: + fc2_b + z_res, transposed store
            float val = v + p0[gn] + aux[(size_t)gm * ldc + gn];
            int b = gm / NVAR;
            C[(size_t)b * (PRED * NVAR) + (size_t)gn * NVAR + d] = val;
        }
    }
}

// ---------------------------------------------------------------------------
// Prep kernels
// ---------------------------------------------------------------------------
__global__ __launch_bounds__(256)
void k_dctmat(float* __restrict__ Co, float* __restrict__ CoTd, float* __restrict__ dvec)
{
    int idx = blockIdx.x * blockDim.x + threadIdx.x;
    if (idx >= LSEQ * LSEQ) return;
    int o = idx >> 9, l = idx & 511;
    float ang = 3.14159265358979323846f * (2.0f * (float)l + 1.0f) * (float)o
                / (2.0f * (float)LSEQ);
    float ck  = (o == 0) ? rsqrtf((float)LSEQ) : sqrtf(2.0f / (float)LSEQ);
    float co  = cosf(ang) * ck;
    float dv  = (o == 0) ? 2.0f * sqrtf((float)LSEQ) : sqrtf(2.0f * (float)LSEQ);
    Co[o * LSEQ + l]   = co;
    CoTd[l * LSEQ + o] = co * dv;
    if (l == 0) dvec[o] = dv;
}

// W2[(n*16+p)*96+o] = sum_j embed_w[p,j] * lin_res_w[n*256+j, o]
__global__ __launch_bounds__(256)
void k_prep_w2(const float* __restrict__ ew, const float* __restrict__ lw,
               float* __restrict__ W2)
{
    int idx = blockIdx.x * blockDim.x + threadIdx.x;
    if (idx >= LSEQ * PRED) return;
    int row = idx / PRED, o = idx % PRED;
    int n = row >> 4, p = row & 15;
    float s = 0.f;
    for (int j = 0; j < 256; ++j)
        s += ew[p * 256 + j] * lw[(size_t)(n * 256 + j) * PRED + o];
    W2[idx] = s;
}

// blin[o] = lin_res_b[o] + sum_{n,j} embed_b[j] * lin_res_w[n*256+j, o]
__global__ void k_prep_blin(const float* __restrict__ eb, const float* __restrict__ lw,
                            const float* __restrict__ lb, float* __restrict__ blin)
{
    int o = threadIdx.x;
    if (o >= PRED) return;
    float s = lb[o];
    for (int n = 0; n < NPAT; ++n)
        for (int j = 0; j < 256; ++j)
            s += eb[j] * lw[(size_t)(n * 256 + j) * PRED + o];
    blin[o] = s;
}

// E2[q,p] = embed_w @ depth_res_w ; bias2[p] = embed_b @ depth_res_w + depth_res_b
__global__ void k_prep_e2(const float* __restrict__ ew, const float* __restrict__ eb,
                          const float* __restrict__ drw, const float* __restrict__ drb,
                          float* __restrict__ E2, float* __restrict__ b2)
{
    int t = threadIdx.x;
    if (t < 256) {
        int q = t >> 4, p = t & 15;
        float s = 0.f;
        for (int j = 0; j < 256; ++j) s += ew[q * 256 + j] * drw[j * 16 + p];
        E2[t] = s;
    }
    if (t < 16) {
        float s = drb[t];
        for (int j = 0; j < 256; ++j) s += eb[j] * drw[j * 16 + t];
        b2[t] = s;
    }
}

// E3[n,q,p] = sum_j embed_w[q, p*16+j] * dc_w[n,j]
// b3[n,p]   = sum_j embed_b[p*16+j]   * dc_w[n,j] + dc_b[n]
__global__ __launch_bounds__(256)
void k_prep_e3(const float* __restrict__ ew, const float* __restrict__ eb,
               const float* __restrict__ dcw, const float* __restrict__ dcb,
               float* __restrict__ E3, float* __restrict__ b3)
{
    int idx = blockIdx.x * blockDim.x + threadIdx.x;
    if (idx < NPAT * 256) {
        int n = idx >> 8, q = (idx >> 4) & 15, p = idx & 15;
        float s = 0.f;
        for (int j = 0; j < 16; ++j) s += ew[q * 256 + p * 16 + j] * dcw[n * 16 + j];
        E3[idx] = s;
    }
    if (idx < NPAT * 16) {
        int n = idx >> 4, p = idx & 15;
        float s = dcb[n];
        for (int j = 0; j < 16; ++j) s += eb[p * 16 + j] * dcw[n * 16 + j];
        b3[idx] = s;
    }
}

// ---------------------------------------------------------------------------
// Pipeline kernels
// ---------------------------------------------------------------------------
__global__ __launch_bounds__(256)
void k_transpose(const float* __restrict__ x, float* __restrict__ xt)
{
    int idx = blockIdx.x * blockDim.x + threadIdx.x;
    if (idx >= STOT) return;
    int l  = idx & (LSEQ - 1);
    int bd = idx >> 9;
    int d  = bd % NVAR, b = bd / NVAR;
    xt[idx] = x[((size_t)b * LSEQ + l) * NVAR + d];
}

__global__ __launch_bounds__(256)
void k_energy(const float* __restrict__ zdct, float* __restrict__ energy)
{
    int m = blockIdx.x * blockDim.x + threadIdx.x;
    if (m >= BD) return;
    const float* row = zdct + (size_t)m * LSEQ;
    float s = 0.f;
    for (int k = 0; k < LSEQ; ++k) { float v = row[k]; s += v * v; }
    energy[m] = s;
}

// exact median (D=321 odd -> order statistic 160) via stable rank counting
__global__ __launch_bounds__(352)
void k_median(const float* __restrict__ energy, float* __restrict__ med)
{
    __shared__ float se[352];
    int b = blockIdx.x, t = threadIdx.x;
    if (t < NVAR) se[t] = energy[b * NVAR + t];
    __syncthreads();
    if (t < NVAR) {
        float v = se[t];
        int rank = 0;
        for (int j = 0; j < NVAR; ++j) {
            float u = se[j];
            rank += (u < v) || (u == v && j < t);
        }
        if (rank == (NVAR >> 1)) med[b] = v;
    }
}

__global__ __launch_bounds__(256)
void k_ne(const float* __restrict__ energy, const float* __restrict__ med,
          float* __restrict__ ne)
{
    int m = blockIdx.x * blockDim.x + threadIdx.x;
    if (m >= BD) return;
    ne[m] = energy[m] / (med[m / NVAR] + 1e-6f);
}

// quantile order statistics via stable rank counting over all BD elements
__global__ __launch_bounds__(256)
void k_rank(const float* __restrict__ ne, const float* __restrict__ thresh,
            float* __restrict__ thrb)
{
    int t = blockIdx.x * blockDim.x + threadIdx.x;
    if (t >= BD) return;
    float q   = fminf(fmaxf(thresh[0], 0.f), 1.f);
    float pos = q * (float)(BD - 1);
    int lo = (int)floorf(pos);
    int hi = (int)ceilf(pos);
    if (hi > BD - 1) hi = BD - 1;
    float v = ne[t];
    int rank = 0;
    for (int j = 0; j < BD; ++j) {
        float u = ne[j];
        rank += (u < v) || (u == v && j < t);
    }
    if (rank == lo) thrb[0] = v;
    if (rank == hi) thrb[1] = v;
}

__global__ void k_thr(const float* __restrict__ thresh, float* __restrict__ thrb)
{
    float q   = fminf(fmaxf(thresh[0], 0.f), 1.f);
    float pos = q * (float)(BD - 1);
    float lo  = floorf(pos);
    thrb[2] = thrb[0] + (pos - lo) * (thrb[1] - thrb[0]);
}

// z_scaled = gelu(conv1(z_dct * mask)) / dvec   (in place on zdct buffer)
__global__ __launch_bounds__(256)
void k_mask(float* __restrict__ zdct, const float* __restrict__ ne,
            const float* __restrict__ thrb, const float* __restrict__ c1w,
            const float* __restrict__ c1b, const float* __restrict__ dvec)
{
    int idx = blockIdx.x * blockDim.x + threadIdx.x;
    if (idx >= STOT) return;
    int m = idx >> 9, k = idx & (LSEQ - 1);
    int d = m % NVAR;
    float thr = thrb[2];
    float v = (ne[m] > thr) ? zdct[idx] : 0.f;
    v = geluf(c1w[d] * v + c1b[d]);
    zdct[idx] = v / dvec[k];
}

// folded patch branch: z2 = gelu(patchBN(xt_patch @ (E2 + E3[n]) + biases))
__global__ __launch_bounds__(256)
void k_patch(const float* __restrict__ xt, float* __restrict__ z2,
             const float* __restrict__ E2, const float* __restrict__ b2,
             const float* __restrict__ E3, const float* __restrict__ b3,
             const float* __restrict__ png, const float* __restrict__ pnb,
             const float* __restrict__ pnm, const float* __restrict__ pnv)
{
    int idx = blockIdx.x * blockDim.x + threadIdx.x;
    if (idx >= STOT) return;
    int bd = idx >> 9, off = idx & (LSEQ - 1);
    int n = off >> 4, p = off & 15;
    const float* xrow = xt + (size_t)bd * LSEQ + (n << 4);
    float s = b2[p] + b3[n * 16 + p];
    for (int q = 0; q < 16; ++q)
        s += xrow[q] * (E2[q * 16 + p] + E3[((n * 16 + q) << 4) + p]);
    s = (s - pnm[n]) * (png[n] * rsqrtf(pnv[n] + EPSBN)) + pnb[n];
    z2[idx] = geluf(s);
}

// z_r = sigmoid(conv1(bnD(gelu(dwconv5(z1) + conv2_b))))
__global__ __launch_bounds__(256)
void k_zr(const float* __restrict__ z1, float* __restrict__ zr,
          const float* __restrict__ c2w, const float* __restrict__ c2b,
          const float* __restrict__ c1w, const float* __restrict__ c1b,
          const float* __restrict__ dng, const float* __restrict__ dnb,
          const float* __restrict__ dnm, const float* __restrict__ dnv)
{
    int idx = blockIdx.x * blockDim.x + threadIdx.x;
    if (idx >= STOT) return;
    int m = idx >> 9, l = idx & (LSEQ - 1);
    int d = m % NVAR;
    const float* row = z1 + (size_t)m * LSEQ;
    float s = c2b[d];
    for (int i = 0; i < 5; ++i) {
        int ll = l - 2 + i;
        if (ll >= 0 && ll < LSEQ) s += row[ll] * c2w[d * 5 + i];
    }
    float g = geluf(s);
    float u = c1w[d] * bn_d(g, d, dng, dnb, dnm, dnv) + c1b[d];
    zr[idx] = 1.f / (1.f + expf(-u));
}

// z1c = bnD(gelu(z2_f * z_r)) + gelu(bnD(dwconv7(z1) + conv3_b))
__global__ __launch_bounds__(256)
void k_combine(const float* __restrict__ z1, const float* __restrict__ z2f,
               const float* __restrict__ zr, float* __restrict__ z1c,
               const float* __restrict__ c3w, const float* __restrict__ c3b,
               const float* __restrict__ dng, const float* __restrict__ dnb,
               const float* __restrict__ dnm, const float* __restrict__ dnv)
{
    int idx = blockIdx.x * blockDim.x + threadIdx.x;
    if (idx >= STOT) return;
    int m = idx >> 9, l = idx & (LSEQ - 1);
    int d = m % NVAR;
    const float* row = z1 + (size_t)m * LSEQ;
    float s7 = c3b[d];
    for (int i = 0; i < 7; ++i) {
        int ll = l - 3 + i;
        if (ll >= 0 && ll < LSEQ) s7 += row[ll] * c3w[d * 7 + i];
    }
    float z1b = geluf(bn_d(s7, d, dng, dnb, dnm, dnv));
    float zg  = geluf(z2f[idx] * zr[idx]);
    z1c[idx]  = bn_d(zg, d, dng, dnb, dnm, dnv) + z1b;
}

// DctAtt pre-softmax scalar per (b,d): 5 low-freq ortho-DCT comps of z1c*z2
__global__ __launch_bounds__(256)
void k_att(const float* __restrict__ z1c, const float* __restrict__ z2,
           const float* __restrict__ Co, float* __restrict__ zd,
           const float* __restrict__ adw, const float* __restrict__ adb,
           const float* __restrict__ acw, const float* __restrict__ acb,
           const float* __restrict__ ag, const float* __restrict__ ab,
           const float* __restrict__ am, const float* __restrict__ av)
{
    int m = blockIdx.x * blockDim.x + threadIdx.x;
    if (m >= BD) return;
    const float* a = z1c + (size_t)m * LSEQ;
    const float* b = z2  + (size_t)m * LSEQ;
    float low[5] = {0.f, 0.f, 0.f, 0.f, 0.f};
    for (int l = 0; l < LSEQ; ++l) {
        float p = a[l] * b[l];
#pragma unroll
        for (int j = 0; j < 5; ++j) low[j] += p * Co[j * LSEQ + l];
    }
    float v = adb[0];
#pragma unroll
    for (int j = 0; j < 5; ++j) v += low[j] * adw[j];
    v = geluf(v);
    v = (v - am[0]) * (ag[0] * rsqrtf(av[0] + EPSBN)) + ab[0];
    zd[m] = v * acw[0] + acb[0];
}

__global__ __launch_bounds__(512)
void k_softmax(const float* __restrict__ zd, float* __restrict__ att)
{
    __shared__ float sh[512];
    int b = blockIdx.x, t = threadIdx.x;
    float v = (t < NVAR) ? zd[b * NVAR + t] : -3.0e38f;
    sh[t] = v;
    __syncthreads();
    for (int s = 256; s > 0; s >>= 1) {
        if (t < s) sh[t] = fmaxf(sh[t], sh[t + s]);
        __syncthreads();
    }
    float mx = sh[0];
    __syncthreads();
    float e = (t < NVAR) ? expf(v - mx) : 0.f;
    sh[t] = e;
    __syncthreads();
    for (int s = 256; s > 0; s >>= 1) {
        if (t < s) sh[t] += sh[t + s];
        __syncthreads();
    }
    if (t < NVAR) att[b * NVAR + t] = e / sh[0];
}

__global__ __launch_bounds__(256)
void k_zf(const float* __restrict__ z1c, const float* __restrict__ z2,
          const float* __restrict__ att, float* __restrict__ zf)
{
    int idx = blockIdx.x * blockDim.x + threadIdx.x;
    if (idx >= STOT) return;
    float a1 = att[idx >> 9];
    zf[idx] = z1c[idx] * a1 + z2[idx] * (1.f - a1);
}

// ---------------------------------------------------------------------------
// Launcher
// ---------------------------------------------------------------------------
extern "C" void kernel_launch(void* const* d_in, const int* in_sizes, int n_in,
                              void* d_out, int out_size, void* d_ws, size_t ws_size,
                              hipStream_t stream)
{
    (void)in_sizes; (void)n_in; (void)out_size; (void)ws_size;

    const float* x    = (const float*)d_in[0];
    const float* c1w  = (const float*)d_in[1];
    const float* c1b  = (const float*)d_in[2];
    const float* c2w  = (const float*)d_in[3];
    const float* c2b  = (const float*)d_in[4];
    const float* c3w  = (const float*)d_in[5];
    const float* c3b  = (const float*)d_in[6];
    const float* dng  = (const float*)d_in[7];
    const float* dnb  = (const float*)d_in[8];
    const float* dnm  = (const float*)d_in[9];
    const float* dnv  = (const float*)d_in[10];
    const float* png  = (const float*)d_in[11];
    const float* pnb  = (const float*)d_in[12];
    const float* pnm  = (const float*)d_in[13];
    const float* pnv  = (const float*)d_in[14];
    const float* ew   = (const float*)d_in[15];
    const float* eb   = (const float*)d_in[16];
    const float* lrw  = (const float*)d_in[17];
    const float* lrb  = (const float*)d_in[18];
    const float* drw  = (const float*)d_in[19];
    const float* drb  = (const float*)d_in[20];
    const float* dcw  = (const float*)d_in[21];
    const float* dcb  = (const float*)d_in[22];
    const float* thresh = (const float*)d_in[23];
    const float* adw  = (const float*)d_in[24];
    const float* adb  = (const float*)d_in[25];
    const float* acw  = (const float*)d_in[26];
    const float* acb  = (const float*)d_in[27];
    const float* ag   = (const float*)d_in[28];
    const float* ab   = (const float*)d_in[29];
    const float* am   = (const float*)d_in[30];
    const float* av   = (const float*)d_in[31];
    const float* f1w  = (const float*)d_in[32];
    const float* f1b  = (const float*)d_in[33];
    const float* f2w  = (const float*)d_in[34];
    const float* f2b  = (const float*)d_in[35];
    float* out = (float*)d_out;

    // workspace layout (floats)
    float* w    = (float*)d_ws;
    float* xt   = w + 0 * (size_t)STOT;   // later reused for z2_f
    float* zdct = w + 1 * (size_t)STOT;   // later z_scaled (in place)
    float* z1   = w + 2 * (size_t)STOT;
    float* z2   = w + 3 * (size_t)STOT;
    float* zr   = w + 4 * (size_t)STOT;   // later reused for zf
    float* z1c  = w + 5 * (size_t)STOT;
    float* sm   = w + 6 * (size_t)STOT;
    float* Co   = sm;  sm += LSEQ * LSEQ;
    float* CoTd = sm;  sm += LSEQ * LSEQ;
    float* dvec = sm;  sm += LSEQ;
    float* W2   = sm;  sm += LSEQ * PRED;
    float* blin = sm;  sm += PRED;
    float* E2   = sm;  sm += 256;
    float* b2   = sm;  sm += 16;
    float* E3   = sm;  sm += NPAT * 256;
    float* b3   = sm;  sm += NPAT * 16;
    float* energy = sm; sm += BD;
    float* nebuf  = sm; sm += BD;
    float* med    = sm; sm += NB;
    float* thrb   = sm; sm += 8;
    float* zdb    = sm; sm += BD;
    float* attb   = sm; sm += BD;
    float* zres   = sm; sm += (size_t)BD * PRED;
    float* hbuf   = sm; sm += (size_t)BD * PRED;

    const int EB = 256;
    const int GS = STOT / EB;                 // 20544, exact
    const int GM = (BD + EB - 1) / EB;        // 41
    dim3 gDCT(BD / 32, LSEQ / 32);            // (321,16)
    dim3 gN96(BD / 32, PRED / 32);            // (321,3)

    // prep
    k_dctmat  <<<(LSEQ * LSEQ) / EB, EB, 0, stream>>>(Co, CoTd, dvec);
    k_prep_w2 <<<(LSEQ * PRED) / EB, EB, 0, stream>>>(ew, lrw, W2);
    k_prep_blin<<<1, PRED, 0, stream>>>(eb, lrw, lrb, blin);
    k_prep_e2 <<<1, 256, 0, stream>>>(ew, eb, drw, drb, E2, b2);
    k_prep_e3 <<<NPAT, 256, 0, stream>>>(ew, eb, dcw, dcb, E3, b3);

    // xt = transpose(x); z_dct = xt @ Co^T * dvec
    k_transpose<<<GS, EB, 0, stream>>>(x, xt);
    wmma_gemm_f32<<<gDCT, 128, 0, stream>>>(xt, CoTd, zdct, LSEQ, LSEQ, LSEQ, LSEQ,
                                            0, nullptr, nullptr, nullptr,
                                            nullptr, nullptr, nullptr, nullptr);

    // adaptive mask threshold
    k_energy<<<GM, EB, 0, stream>>>(zdct, energy);
    k_median<<<NB, 352, 0, stream>>>(energy, med);
    k_ne    <<<GM, EB, 0, stream>>>(energy, med, nebuf);
    k_rank  <<<GM, EB, 0, stream>>>(nebuf, thresh, thrb);
    k_thr   <<<1, 1, 0, stream>>>(thresh, thrb);
    k_mask  <<<GS, EB, 0, stream>>>(zdct, nebuf, thrb, c1w, c1b, dvec);

    // z1 = z_scaled @ Co + freq_res(xt)
    wmma_gemm_f32<<<gDCT, 128, 0, stream>>>(zdct, Co, z1, LSEQ, LSEQ, LSEQ, LSEQ,
                                            1, xt, c1w, c1b,
                                            nullptr, nullptr, nullptr, nullptr);

    // z_res = xt @ W2 + blin   (folded patch-embed + linear residual)
    wmma_gemm_f32<<<gN96, 128, 0, stream>>>(xt, W2, zres, LSEQ, LSEQ, PRED, PRED,
                                            3, nullptr, blin, nullptr,
                                            nullptr, nullptr, nullptr, nullptr);

    // z2 via folded per-patch 16x16 operators
    k_patch<<<GS, EB, 0, stream>>>(xt, z2, E2, b2, E3, b3, png, pnb, pnm, pnv);

    // z2_f = conv1(bnD(gelu(z2 @ Co^T * dvec)))  (stored into xt buffer)
    wmma_gemm_f32<<<gDCT, 128, 0, stream>>>(z2, CoTd, xt, LSEQ, LSEQ, LSEQ, LSEQ,
                                            2, nullptr, c1w, c1b, dng, dnb, dnm, dnv);

    // gating + combine
    k_zr<<<GS, EB, 0, stream>>>(z1, zr, c2w, c2b, c1w, c1b, dng, dnb, dnm, dnv);
    k_combine<<<GS, EB, 0, stream>>>(z1, xt, zr, z1c, c3w, c3b, dng, dnb, dnm, dnv);

    // DCT attention
    k_att<<<GM, EB, 0, stream>>>(z1c, z2, Co, zdb, adw, adb, acw, acb, ag, ab, am, av);
    k_softmax<<<NB, 512, 0, stream>>>(zdb, attb);
    k_zf<<<GS, EB, 0, stream>>>(z1c, z2, attb, zr);      // zf into zr buffer

    // MLP head: h = gelu(zf @ fc1 + b1); out = h @ fc2 + b2 + z_res (transposed)
    wmma_gemm_f32<<<gN96, 128, 0, stream>>>(zr, f1w, hbuf, LSEQ, LSEQ, PRED, PRED,
                                            4, nullptr, f1b, nullptr,
                                            nullptr, nullptr, nullptr, nullptr);
    wmma_gemm_f32<<<gN96, 128, 0, stream>>>(hbuf, f2w, out, PRED, PRED, PRED, PRED,
                                            5, zres, f2b, nullptr,
                                            nullptr, nullptr, nullptr, nullptr);
}